// VectorQuantizer_49194555408575
// MI455X (gfx1250) — compile-verified
//
#include <hip/hip_runtime.h>

// VQ-VAE vector quantizer for MI455X (gfx1250).
// GEMM part (131072 x 1024 x 64, fp32) done with V_WMMA_F32_16X16X4_F32.
// argmin(||x||^2 - 2x.e + ||e||^2)  ==  argmax(x.e - 0.5||e||^2); the bias
// -0.5||e||^2 is preloaded into the WMMA C accumulator.

typedef __attribute__((ext_vector_type(2))) float v2f;
typedef __attribute__((ext_vector_type(8))) float v8f;

#define NUM_CODES 1024
#define DIM       64
#define HW        4096            // 64*64
#define CHW       (DIM * HW)      // 262144
#define NROWS     131072          // 32*64*64
#define QELEMS    8388608         // 32*64*64*64

// -------- bias[n] = -0.5 * ||codebook[n]||^2 --------------------------------
__global__ void vq_bias_kernel(const float* __restrict__ codebook,
                               float* __restrict__ bias) {
    int n = blockIdx.x * blockDim.x + threadIdx.x;
    if (n < NUM_CODES) {
        const float* row = codebook + n * DIM;
        float s = 0.f;
        #pragma unroll
        for (int c = 0; c < DIM; ++c) s += row[c] * row[c];
        bias[n] = -0.5f * s;
    }
}

// -------- main fused kernel: WMMA scores -> argmax -> gather -> ST out ------
__launch_bounds__(256)
__global__ void vq_main_kernel(const float* __restrict__ x,
                               const float* __restrict__ codebook,
                               const float* __restrict__ bias,
                               float* __restrict__ out_q,
                               float* __restrict__ out_idx,
                               float* __restrict__ block_partials) {
    __shared__ float s_wave[8];

    const int lane        = threadIdx.x & 31;
    const int waveInBlock = threadIdx.x >> 5;
    const int waveId      = blockIdx.x * 8 + waveInBlock;
    const int rowBase     = waveId * 16;          // 16 x-vectors per wave
    const int b           = rowBase >> 12;        // batch index (rowBase/4096)
    const int hwBase      = rowBase & 4095;       // never crosses a batch (16 | 4096)
    const int m           = lane & 15;            // M row within tile
    const int hiHalf      = lane >> 4;            // 0 or 1 (K quad offset)

    // ---- Load A tile (16 rows x 64 channels of x) in WMMA f32 16x16x4 layout.
    // A reg j holds (row=m, channel c = 4*(j>>1) + (j&1) + 2*hiHalf).
    float a[32];
    const float* xb = x + (size_t)b * CHW + hwBase + m;
    #pragma unroll
    for (int j = 0; j < 32; ++j) {
        int c = 4 * (j >> 1) + (j & 1) + 2 * hiHalf;
        a[j] = xb[c * HW];
    }

    // Per-lane B gather base: B[k][n] = codebook[n][k]; lane holds column n = tile*16+m,
    // K offset 2*hiHalf (ISA B layout: VGPR0 = K {0|2}, VGPR1 = K {1|3}).
    const float* bbase = codebook + m * DIM + 2 * hiHalf;

    float bestScore[8];
    int   bestIdx[8];
    #pragma unroll
    for (int v = 0; v < 8; ++v) { bestScore[v] = -3.0e38f; bestIdx[v] = 0; }

    for (int tile = 0; tile < 64; ++tile) {
        const float biasv  = bias[tile * 16 + m];       // -0.5||e_n||^2 for this lane's column
        const float* bptr  = bbase + tile * 16 * DIM;

        v2f bvec[16];
        #pragma unroll
        for (int ks = 0; ks < 16; ++ks)
            bvec[ks] = *(const v2f*)(bptr + 4 * ks);    // global_load_b64 gather (L2-hot)

        v8f acc;
        #pragma unroll
        for (int v = 0; v < 8; ++v) acc[v] = biasv;     // C preloaded with bias

        #pragma unroll
        for (int ks = 0; ks < 16; ++ks) {
            v2f av;
            av.x = a[2 * ks];
            av.y = a[2 * ks + 1];
            acc = __builtin_amdgcn_wmma_f32_16x16x4_f32(
                false, av, false, bvec[ks], (short)0, acc, false, false);
        }

        const int n = tile * 16 + m;                    // this lane's column index
        #pragma unroll
        for (int v = 0; v < 8; ++v) {
            if (acc[v] > bestScore[v]) { bestScore[v] = acc[v]; bestIdx[v] = n; }
        }
    }

    // ---- reduce argmax across each 16-lane half (D layout: lanes0-15 row v,
    // lanes16-31 row v+8). Tie-break: smaller index (matches argmin-first).
    #pragma unroll
    for (int v = 0; v < 8; ++v) {
        float s = bestScore[v];
        int   i = bestIdx[v];
        #pragma unroll
        for (int mask = 1; mask < 16; mask <<= 1) {
            float so = __shfl_xor(s, mask, 32);
            int   io = __shfl_xor(i, mask, 32);
            if (so > s || (so == s && io < i)) { s = so; i = io; }
        }
        bestScore[v] = s;
        bestIdx[v]   = i;
    }

    // ---- broadcast: each lane picks the winning index of its row m.
    int myIdx = 0;
    #pragma unroll
    for (int v = 0; v < 8; ++v) {
        int r0 = __shfl(bestIdx[v], 0, 32);    // row v      (lanes 0-15 group)
        int r1 = __shfl(bestIdx[v], 16, 32);   // row v + 8  (lanes 16-31 group)
        if (m == v)     myIdx = r0;
        if (m == v + 8) myIdx = r1;
    }

    if (lane < 16) out_idx[rowBase + lane] = (float)myIdx;

    // ---- gather codebook row, write straight-through output, accumulate loss.
    const float* crow = codebook + myIdx * DIM;
    float* qb = out_q + (size_t)b * CHW + hwBase + m;
    float lossAcc = 0.f;
    #pragma unroll
    for (int j = 0; j < 32; ++j) {
        int c   = 4 * (j >> 1) + (j & 1) + 2 * hiHalf;
        float q = crow[c];
        float d = q - a[j];                   // quantized - x
        lossAcc += d * d;
        qb[c * HW] = a[j] + d;                // x + (q - x), as in the reference
    }

    // ---- deterministic loss partial: wave shuffle-reduce -> LDS -> block sum.
    #pragma unroll
    for (int mask = 1; mask < 32; mask <<= 1)
        lossAcc += __shfl_xor(lossAcc, mask, 32);
    if (lane == 0) s_wave[waveInBlock] = lossAcc;
    __syncthreads();
    if (threadIdx.x == 0) {
        float s = 0.f;
        #pragma unroll
        for (int w = 0; w < 8; ++w) s += s_wave[w];
        block_partials[blockIdx.x] = s;
    }
}

// -------- final deterministic loss reduction: 1024 partials -> scalar -------
__global__ void vq_loss_kernel(const float* __restrict__ partials,
                               float* __restrict__ out_loss) {
    __shared__ float sm[256];
    int t = threadIdx.x;
    float s = partials[t] + partials[t + 256] + partials[t + 512] + partials[t + 768];
    sm[t] = s;
    __syncthreads();
    for (int off = 128; off > 0; off >>= 1) {
        if (t < off) sm[t] += sm[t + off];
        __syncthreads();
    }
    if (t == 0)
        out_loss[0] = sm[0] * (1.25f / (float)QELEMS);  // (1 + 0.25) * MSE
}

extern "C" void kernel_launch(void* const* d_in, const int* in_sizes, int n_in,
                              void* d_out, int out_size, void* d_ws, size_t ws_size,
                              hipStream_t stream) {
    const float* x        = (const float*)d_in[0];  // [32,64,64,64] f32
    const float* codebook = (const float*)d_in[1];  // [1024,64] f32

    float* out_q    = (float*)d_out;          // [32,64,64,64] quantized_st
    float* out_loss = out_q + QELEMS;         // [1] loss
    float* out_idx  = out_loss + 1;           // [32,64,64] indices (as float)

    float* partials = (float*)d_ws;           // 1024 floats: per-block loss partials
    float* bias     = partials + 1024;        // 1024 floats: -0.5||e||^2

    vq_bias_kernel<<<NUM_CODES / 256, 256, 0, stream>>>(codebook, bias);
    vq_main_kernel<<<NROWS / (16 * 8), 256, 0, stream>>>(x, codebook, bias,
                                                         out_q, out_idx, partials);
    vq_loss_kernel<<<1, 256, 0, stream>>>(partials, out_loss);
}